// FP_Module_angle_label_18433999634713
// MI455X (gfx1250) — compile-verified
//
#include <hip/hip_runtime.h>
#include <hip/hip_bf16.h>

typedef __attribute__((ext_vector_type(2))) float        v2f;
typedef __attribute__((ext_vector_type(8))) float        v8f;
typedef __attribute__((ext_vector_type(4))) unsigned int v4u;
typedef __attribute__((ext_vector_type(4))) int          v4i;
typedef __attribute__((ext_vector_type(8))) int          v8i;

#define B_    16
#define N1_   4096
#define N2_   1024
#define C2_   256
#define L_    16
#define K_    3
#define CIN_  304      // 256 + 32 + 16
#define M0_   256
#define M1_   128
#define ROWS_ (B_ * N1_ * K_)   // 196608
#define BN_EPS_ 1e-5f

// ---------------------------------------------------------------------------
// TDM: 2-D tensor_load_to_lds.  Tile = tile_d1 rows x tile_d0 DWORDs, row
// stride in global = stride0 DWORDs.  LDS padding: 1 DWORD inserted after
// every (1<<pad_code) 8-byte units (pad_code=2 -> after each 32B row),
// producing an odd LDS pitch of tile_d0+1 DWORDs (bank-conflict-free).
// ---------------------------------------------------------------------------
static __device__ __forceinline__ void tdm_load_2d(unsigned lds_addr,
                                                   const void* gptr,
                                                   unsigned tensor_d0,
                                                   unsigned tensor_d1,
                                                   unsigned tile_d0,
                                                   unsigned tile_d1,
                                                   unsigned stride0,
                                                   unsigned pad_code) {
    unsigned long long ga = (unsigned long long)gptr;
    v4u g0;
    g0[0] = 1u;                                           // count=1, user desc
    g0[1] = lds_addr;                                     // bits[63:32]
    g0[2] = (unsigned)ga;                                 // global_addr lo
    g0[3] = (unsigned)((ga >> 32) & 0x01FFFFFFu) | (2u << 30);  // addr hi | type=2
    v8i g1;
    g1[0] = (int)((2u << 16) | (1u << 20) | (pad_code << 22));  // data_size=4B, pad_en, pad_amt=1dw
    g1[1] = (int)((tensor_d0 & 0xFFFFu) << 16);           // tensor_dim0 lo16 @bit48
    g1[2] = (int)((tensor_d0 >> 16) | ((tensor_d1 & 0xFFFFu) << 16));
    g1[3] = (int)((tensor_d1 >> 16) | (tile_d0 << 16));   // tile_dim0 @bit112
    g1[4] = (int)(tile_d1 & 0xFFFFu);                     // tile_dim1, tile_dim2=0
    g1[5] = (int)stride0;                                 // tensor_dim0_stride lo32
    g1[6] = 0;
    g1[7] = 0;
    v4i z4 = {0, 0, 0, 0};
#if defined(__clang_major__) && (__clang_major__ >= 23)
    v8i z8 = {0, 0, 0, 0, 0, 0, 0, 0};
    __builtin_amdgcn_tensor_load_to_lds(g0, g1, z4, z4, z8, 0);
#else
    __builtin_amdgcn_tensor_load_to_lds(g0, g1, z4, z4, 0);
#endif
}

// ---------------------------------------------------------------------------
// Kernel 1: exact-formula KNN (top-3) per (b, n1) row
// ---------------------------------------------------------------------------
__global__ void __launch_bounds__(256) knn_kernel(const float* __restrict__ xyz1,
                                                  const float* __restrict__ xyz2,
                                                  int* __restrict__ inds) {
    __shared__ float sx[N2_ * 3];
    __shared__ float ss2[N2_];
    const int tid = threadIdx.x;
    const int b   = blockIdx.x / (N1_ / 256);
    const int n1  = (blockIdx.x % (N1_ / 256)) * 256 + tid;

    for (int i = tid; i < N2_ * 3; i += 256) sx[i] = xyz2[b * N2_ * 3 + i];
    __syncthreads();
    for (int j = tid; j < N2_; j += 256) {
        float x = sx[j * 3 + 0], y = sx[j * 3 + 1], z = sx[j * 3 + 2];
        ss2[j] = x * x + y * y + z * z;
    }
    __syncthreads();

    const float px = xyz1[(b * N1_ + n1) * 3 + 0];
    const float py = xyz1[(b * N1_ + n1) * 3 + 1];
    const float pz = xyz1[(b * N1_ + n1) * 3 + 2];
    const float s1 = px * px + py * py + pz * pz;

    float d0 = 3.0e38f, d1 = 3.0e38f, d2v = 3.0e38f;
    int   i0 = 0, i1 = 0, i2 = 0;
    for (int j = 0; j < N2_; ++j) {
        float dot = px * sx[j * 3 + 0] + py * sx[j * 3 + 1] + pz * sx[j * 3 + 2];
        float d = s1 + ss2[j] - 2.0f * dot;
        if (d < 0.0f) d = 1e-7f;
        if (d < d0)       { d2v = d1; i2 = i1; d1 = d0; i1 = i0; d0 = d; i0 = j; }
        else if (d < d1)  { d2v = d1; i2 = i1; d1 = d;  i1 = j; }
        else if (d < d2v) { d2v = d;  i2 = j; }
    }
    const int base = (b * N1_ + n1) * K_;
    inds[base + 0] = i0;
    inds[base + 1] = i1;
    inds[base + 2] = i2;
}

// ---------------------------------------------------------------------------
// Kernel 2: relpos (7-dim) -> pos_pre (32 ch) + channel sum/sumsq
// ---------------------------------------------------------------------------
__global__ void __launch_bounds__(256) relpos_kernel(const float* __restrict__ points1,
                                                     const float* __restrict__ xyz2,
                                                     const float* __restrict__ dir1,
                                                     const float* __restrict__ dir2,
                                                     const int*   __restrict__ inds,
                                                     const float* __restrict__ w_pos,
                                                     float* __restrict__ pos_pre,
                                                     float* __restrict__ pos_sum,
                                                     float* __restrict__ pos_ssq) {
    __shared__ float wp[32 * 7];
    __shared__ float ssum[32], sssq[32];
    const int tid = threadIdx.x;
    if (tid < 224) wp[tid] = w_pos[tid];
    if (tid < 32) { ssum[tid] = 0.0f; sssq[tid] = 0.0f; }
    __syncthreads();

    const int row = blockIdx.x * 256 + tid;          // < ROWS_ (exact)
    const int b   = row / (N1_ * K_);
    const int rem = row % (N1_ * K_);
    const int n1  = rem / K_;
    const int ind = inds[row];

    const float p0 = points1[(b * N1_ + n1) * 3 + 0];
    const float p1 = points1[(b * N1_ + n1) * 3 + 1];
    const float p2 = points1[(b * N1_ + n1) * 3 + 2];
    const float* c2 = xyz2 + (size_t)(b * N2_ + ind) * 3;
    const float rx = p0 - c2[0], ry = p1 - c2[1], rz = p2 - c2[2];
    const float rn = sqrtf(rx * rx + ry * ry + rz * rz);
    const float iv = 1.0f / fmaxf(rn, 1e-12f);
    const float ux = rx * iv, uy = ry * iv, uz = rz * iv;

    const float* D1 = dir1 + (size_t)(b * N2_ + ind) * 9;
    const float* D2 = dir2 + (size_t)(b * N2_ + ind) * 9;
    float rp[7];
    rp[0] = rn;
    rp[1] = D1[0] * ux + D1[1] * uy + D1[2] * uz;
    rp[2] = D1[3] * ux + D1[4] * uy + D1[5] * uz;
    rp[3] = D1[6] * ux + D1[7] * uy + D1[8] * uz;
    rp[4] = D2[0] * ux + D2[1] * uy + D2[2] * uz;
    rp[5] = D2[3] * ux + D2[4] * uy + D2[5] * uz;
    rp[6] = D2[6] * ux + D2[7] * uy + D2[8] * uz;

    #pragma unroll 4
    for (int o = 0; o < 32; ++o) {
        float v = 0.0f;
        #pragma unroll
        for (int c = 0; c < 7; ++c) v += rp[c] * wp[o * 7 + c];
        pos_pre[(size_t)row * 32 + o] = v;
        atomicAdd(&ssum[o], v);
        atomicAdd(&sssq[o], v * v);
    }
    __syncthreads();
    if (tid < 32) {
        atomicAdd(&pos_sum[tid], ssum[tid]);
        atomicAdd(&pos_ssq[tid], sssq[tid]);
    }
}

// ---------------------------------------------------------------------------
// Kernel 3: fold BN stats into per-channel scale/shift
// ---------------------------------------------------------------------------
__global__ void bn_finalize_kernel(const float* __restrict__ sum,
                                   const float* __restrict__ ssq,
                                   const float* __restrict__ gamma,
                                   const float* __restrict__ beta,
                                   float* __restrict__ scale,
                                   float* __restrict__ shift,
                                   int C, float invCnt) {
    const int c = blockIdx.x * blockDim.x + threadIdx.x;
    if (c < C) {
        float m  = sum[c] * invCnt;
        float v  = ssq[c] * invCnt - m * m;
        float sc = gamma[c] * rsqrtf(v + BN_EPS_);
        scale[c] = sc;
        shift[c] = beta[c] - m * sc;
    }
}

// ---------------------------------------------------------------------------
// Kernel 4: GEMM0 (rows x 304) @ w0^T -> y0 (rows x 256)
// WMMA f32 16x16x4; B tiles staged by TDM (double-buffered, pitch 9)
// ---------------------------------------------------------------------------
__global__ void __launch_bounds__(256) gemm0_kernel(const float* __restrict__ feat2,
                                                    const float* __restrict__ pos_pre,
                                                    const float* __restrict__ label,
                                                    const int*   __restrict__ inds,
                                                    const float* __restrict__ w0,
                                                    const float* __restrict__ pos_scale,
                                                    const float* __restrict__ pos_shift,
                                                    float* __restrict__ y0,
                                                    float* __restrict__ bn0_sum,
                                                    float* __restrict__ bn0_ssq) {
    __shared__ float Ash[32 * 305];
    __shared__ float Bsh[2][256 * 9];
    __shared__ int   rind[32];
    __shared__ float ssum[256], sssq[256];

    const int tid  = threadIdx.x;
    const int row0 = blockIdx.x * 32;          // 32 divides N1_*K_ per batch
    const int b    = row0 / (N1_ * K_);
    if (tid < 32) rind[tid] = inds[row0 + tid];
    ssum[tid] = 0.0f;
    sssq[tid] = 0.0f;
    __syncthreads();

    // Stage A: [feat2 gather | BN+LeakyReLU(pos) | label]
    for (int i = tid; i < 32 * CIN_; i += 256) {
        const int r = i / CIN_, c = i % CIN_;
        float v;
        if (c < C2_) {
            v = feat2[(size_t)(b * N2_ + rind[r]) * C2_ + c];
        } else if (c < C2_ + 32) {
            const int p = c - C2_;
            float x = pos_pre[(size_t)(row0 + r) * 32 + p] * pos_scale[p] + pos_shift[p];
            v = x > 0.0f ? x : 0.2f * x;
        } else {
            v = label[b * L_ + (c - C2_ - 32)];
        }
        Ash[r * 305 + c] = v;
    }

    const int wave = tid >> 5, lane = tid & 31;
    const int ln = lane & 15, hi = lane >> 4;      // hi: K-half select
    const int mtile = wave >> 2;                   // 0..1
    const int n0    = (wave & 3) * 64;             // 4 ntiles per wave
    const int arow  = mtile * 16 + ln;

    const unsigned ldsB0 = (unsigned)(size_t)(void*)&Bsh[0][0];
    const unsigned ldsB1 = (unsigned)(size_t)(void*)&Bsh[1][0];

    v8f acc0 = {}, acc1 = {}, acc2 = {}, acc3 = {};

    // prologue: DMA first weight chunk
    if (wave == 0) {
        tdm_load_2d(ldsB0, w0, CIN_, M0_, 8, M0_, CIN_, 2);
        __builtin_amdgcn_s_wait_tensorcnt(0);
    }
    __syncthreads();

    const int NC = CIN_ / 8;                       // 38 chunks
    for (int ch = 0; ch < NC; ++ch) {
        const int kc  = ch * 8;
        const int cur = ch & 1;
        if (wave == 0 && ch + 1 < NC)              // async DMA of next chunk
            tdm_load_2d(cur ? ldsB0 : ldsB1, w0 + kc + 8, CIN_, M0_, 8, M0_, CIN_, 2);
        const float* Bs = Bsh[cur];
        #pragma unroll
        for (int ks = 0; ks < 8; ks += 4) {
            const int kk = ks + hi * 2;
            v2f a, b0v, b1v, b2v, b3v;
            a.x = Ash[arow * 305 + kc + kk];
            a.y = Ash[arow * 305 + kc + kk + 1];
            b0v.x = Bs[(n0 +  0 + ln) * 9 + kk]; b0v.y = Bs[(n0 +  0 + ln) * 9 + kk + 1];
            b1v.x = Bs[(n0 + 16 + ln) * 9 + kk]; b1v.y = Bs[(n0 + 16 + ln) * 9 + kk + 1];
            b2v.x = Bs[(n0 + 32 + ln) * 9 + kk]; b2v.y = Bs[(n0 + 32 + ln) * 9 + kk + 1];
            b3v.x = Bs[(n0 + 48 + ln) * 9 + kk]; b3v.y = Bs[(n0 + 48 + ln) * 9 + kk + 1];
            acc0 = __builtin_amdgcn_wmma_f32_16x16x4_f32(false, a, false, b0v, (short)0, acc0, false, false);
            acc1 = __builtin_amdgcn_wmma_f32_16x16x4_f32(false, a, false, b1v, (short)0, acc1, false, false);
            acc2 = __builtin_amdgcn_wmma_f32_16x16x4_f32(false, a, false, b2v, (short)0, acc2, false, false);
            acc3 = __builtin_amdgcn_wmma_f32_16x16x4_f32(false, a, false, b3v, (short)0, acc3, false, false);
        }
        if (wave == 0) __builtin_amdgcn_s_wait_tensorcnt(0);
        __syncthreads();
    }

    const int rbase = row0 + mtile * 16 + hi * 8;
    auto emit = [&](const v8f& acc, int nt) {
        const int n = n0 + nt * 16 + ln;
        float s = 0.0f, ss = 0.0f;
        #pragma unroll
        for (int v = 0; v < 8; ++v) {
            float val = acc[v];
            y0[(size_t)(rbase + v) * M0_ + n] = val;
            s += val; ss += val * val;
        }
        atomicAdd(&ssum[n], s);
        atomicAdd(&sssq[n], ss);
    };
    emit(acc0, 0); emit(acc1, 1); emit(acc2, 2); emit(acc3, 3);
    __syncthreads();
    atomicAdd(&bn0_sum[tid], ssum[tid]);
    atomicAdd(&bn0_ssq[tid], sssq[tid]);
}

// ---------------------------------------------------------------------------
// Kernel 5: GEMM1 (rows x 256) @ w1^T -> y1 (rows x 128), BN0+ReLU on the fly
// ---------------------------------------------------------------------------
__global__ void __launch_bounds__(256) gemm1_kernel(const float* __restrict__ y0,
                                                    const float* __restrict__ w1,
                                                    const float* __restrict__ s0,
                                                    const float* __restrict__ sh0,
                                                    float* __restrict__ y1,
                                                    float* __restrict__ bn1_sum,
                                                    float* __restrict__ bn1_ssq) {
    __shared__ float Ash[32 * 257];
    __shared__ float Bsh[2][128 * 9];
    __shared__ float ssum[128], sssq[128];

    const int tid  = threadIdx.x;
    const int row0 = blockIdx.x * 32;
    if (tid < 128) { ssum[tid] = 0.0f; sssq[tid] = 0.0f; }

    for (int i = tid; i < 32 * 256; i += 256) {
        const int r = i >> 8, c = i & 255;
        float x = y0[(size_t)(row0 + r) * M0_ + c] * s0[c] + sh0[c];
        Ash[r * 257 + c] = x > 0.0f ? x : 0.0f;
    }

    const int wave = tid >> 5, lane = tid & 31;
    const int ln = lane & 15, hi = lane >> 4;
    const int mtile = wave >> 2;               // 0..1
    const int n0    = (wave & 3) * 32;         // 2 ntiles per wave
    const int arow  = mtile * 16 + ln;

    const unsigned ldsB0 = (unsigned)(size_t)(void*)&Bsh[0][0];
    const unsigned ldsB1 = (unsigned)(size_t)(void*)&Bsh[1][0];

    v8f acc0 = {}, acc1 = {};

    if (wave == 0) {
        tdm_load_2d(ldsB0, w1, M0_, M1_, 8, M1_, M0_, 2);
        __builtin_amdgcn_s_wait_tensorcnt(0);
    }
    __syncthreads();

    const int NC = M0_ / 8;                    // 32 chunks
    for (int ch = 0; ch < NC; ++ch) {
        const int kc  = ch * 8;
        const int cur = ch & 1;
        if (wave == 0 && ch + 1 < NC)
            tdm_load_2d(cur ? ldsB0 : ldsB1, w1 + kc + 8, M0_, M1_, 8, M1_, M0_, 2);
        const float* Bs = Bsh[cur];
        #pragma unroll
        for (int ks = 0; ks < 8; ks += 4) {
            const int kk = ks + hi * 2;
            v2f a, b0v, b1v;
            a.x = Ash[arow * 257 + kc + kk];
            a.y = Ash[arow * 257 + kc + kk + 1];
            b0v.x = Bs[(n0 +  0 + ln) * 9 + kk]; b0v.y = Bs[(n0 +  0 + ln) * 9 + kk + 1];
            b1v.x = Bs[(n0 + 16 + ln) * 9 + kk]; b1v.y = Bs[(n0 + 16 + ln) * 9 + kk + 1];
            acc0 = __builtin_amdgcn_wmma_f32_16x16x4_f32(false, a, false, b0v, (short)0, acc0, false, false);
            acc1 = __builtin_amdgcn_wmma_f32_16x16x4_f32(false, a, false, b1v, (short)0, acc1, false, false);
        }
        if (wave == 0) __builtin_amdgcn_s_wait_tensorcnt(0);
        __syncthreads();
    }

    const int rbase = row0 + mtile * 16 + hi * 8;
    auto emit = [&](const v8f& acc, int nt) {
        const int n = n0 + nt * 16 + ln;
        float s = 0.0f, ss = 0.0f;
        #pragma unroll
        for (int v = 0; v < 8; ++v) {
            float val = acc[v];
            y1[(size_t)(rbase + v) * M1_ + n] = val;
            s += val; ss += val * val;
        }
        atomicAdd(&ssum[n], s);
        atomicAdd(&sssq[n], ss);
    };
    emit(acc0, 0); emit(acc1, 1);
    __syncthreads();
    if (tid < 128) {
        atomicAdd(&bn1_sum[tid], ssum[tid]);
        atomicAdd(&bn1_ssq[tid], sssq[tid]);
    }
}

// ---------------------------------------------------------------------------
// Kernel 6: BN1 + ReLU + sum over k -> out (B, N1, 128)
// ---------------------------------------------------------------------------
__global__ void __launch_bounds__(256) finalize_kernel(const float* __restrict__ y1,
                                                       const float* __restrict__ s1,
                                                       const float* __restrict__ sh1,
                                                       float* __restrict__ out) {
    const int idx = blockIdx.x * 256 + threadIdx.x;   // exact: B_*N1_*M1_
    const int c   = idx & (M1_ - 1);
    const int pt  = idx >> 7;
    const float sc = s1[c], sf = sh1[c];
    const size_t base = (size_t)pt * K_ * M1_ + c;
    float acc = 0.0f;
    #pragma unroll
    for (int kk = 0; kk < K_; ++kk) {
        float x = y1[base + kk * M1_] * sc + sf;
        acc += x > 0.0f ? x : 0.0f;
    }
    out[idx] = acc;
}

__global__ void zero_kernel(float* __restrict__ p, int n) {
    const int i = blockIdx.x * blockDim.x + threadIdx.x;
    if (i < n) p[i] = 0.0f;
}

// ---------------------------------------------------------------------------
extern "C" void kernel_launch(void* const* d_in, const int* in_sizes, int n_in,
                              void* d_out, int out_size, void* d_ws, size_t ws_size,
                              hipStream_t stream) {
    const float* xyz1    = (const float*)d_in[0];
    const float* xyz2    = (const float*)d_in[1];
    const float* points1 = (const float*)d_in[2];
    const float* feat2   = (const float*)d_in[3];
    const float* dir1    = (const float*)d_in[4];
    const float* dir2    = (const float*)d_in[5];
    const float* label   = (const float*)d_in[6];
    const float* w_pos   = (const float*)d_in[7];
    const float* g_pos   = (const float*)d_in[8];
    const float* b_pos   = (const float*)d_in[9];
    const float* w0      = (const float*)d_in[10];
    const float* g0      = (const float*)d_in[11];
    const float* b0      = (const float*)d_in[12];
    const float* w1      = (const float*)d_in[13];
    const float* g1      = (const float*)d_in[14];
    const float* b1      = (const float*)d_in[15];
    float* out = (float*)d_out;

    char* ws = (char*)d_ws;
    size_t off = 0;
    int*   inds    = (int*)(ws + off);   off += (size_t)ROWS_ * sizeof(int);            // 786,432
    float* pos_pre = (float*)(ws + off); off += (size_t)ROWS_ * 32 * sizeof(float);     // 25.2 MB
    float* y0      = (float*)(ws + off); off += (size_t)ROWS_ * M0_ * sizeof(float);    // 201 MB
    float* y1      = (float*)(ws + off); off += (size_t)ROWS_ * M1_ * sizeof(float);    // 100 MB
    float* stats   = (float*)(ws + off);
    float* pos_sum = stats +   0;  float* pos_ssq = stats +  32;
    float* bn0_sum = stats +  64;  float* bn0_ssq = stats + 320;
    float* bn1_sum = stats + 576;  float* bn1_ssq = stats + 704;
    float* pos_sc  = stats + 832;  float* pos_sh  = stats + 864;
    float* s0      = stats + 896;  float* sh0     = stats + 1152;
    float* s1      = stats + 1408; float* sh1     = stats + 1536;

    const float invCnt = 1.0f / (float)ROWS_;

    zero_kernel<<<4, 256, 0, stream>>>(stats, 832);

    knn_kernel<<<B_ * (N1_ / 256), 256, 0, stream>>>(xyz1, xyz2, inds);

    relpos_kernel<<<ROWS_ / 256, 256, 0, stream>>>(points1, xyz2, dir1, dir2, inds,
                                                   w_pos, pos_pre, pos_sum, pos_ssq);

    bn_finalize_kernel<<<1, 256, 0, stream>>>(pos_sum, pos_ssq, g_pos, b_pos,
                                              pos_sc, pos_sh, 32, invCnt);

    gemm0_kernel<<<ROWS_ / 32, 256, 0, stream>>>(feat2, pos_pre, label, inds, w0,
                                                 pos_sc, pos_sh, y0, bn0_sum, bn0_ssq);

    bn_finalize_kernel<<<1, 256, 0, stream>>>(bn0_sum, bn0_ssq, g0, b0, s0, sh0, M0_, invCnt);

    gemm1_kernel<<<ROWS_ / 32, 256, 0, stream>>>(y0, w1, s0, sh0, y1, bn1_sum, bn1_ssq);

    bn_finalize_kernel<<<1, 256, 0, stream>>>(bn1_sum, bn1_ssq, g1, b1, s1, sh1, M1_, invCnt);

    finalize_kernel<<<(B_ * N1_ * M1_) / 256, 256, 0, stream>>>(y1, s1, sh1, out);
}